// SprofECModel_81887846465712
// MI455X (gfx1250) — compile-verified
//
#include <hip/hip_runtime.h>

typedef __attribute__((ext_vector_type(2))) float v2f;
typedef __attribute__((ext_vector_type(8))) float v8f;

#define B_ROWS 8
#define D_IN   1024
#define H1D    2048
#define H2D    1024
#define OUTD   5120

// C = relu(A @ W + bias) for A:[8,K] (padded to 16 rows), W:[K,N], one 16-col
// tile per block. 4 waves per block split K; partials reduced via LDS.
// TRANSPOSED=1 stores out[k_col][b] (8 floats contiguous) instead of out[b][N].
template <int RELU, int TRANSPOSED>
__global__ __launch_bounds__(128) void gemm_wmma_f32(const float* __restrict__ A,
                                                     const float* __restrict__ W,
                                                     const float* __restrict__ bias,
                                                     float* __restrict__ out,
                                                     int K, int N) {
  const int lane = threadIdx.x & 31;
  // Force wave id scalar so the K loop is uniform (no EXEC-masked loop around WMMA)
  const int wave = __builtin_amdgcn_readfirstlane(threadIdx.x >> 5);
  const int m    = lane & 15;          // A row this lane feeds (16x4 frag)
  const int kh   = (lane >> 4) << 1;   // lanes 0-15: K=0,1 ; lanes 16-31: K=2,3
  const int n    = (blockIdx.x << 4) + (lane & 15);

  // zero-mask padding rows 8..15 without diverging EXEC
  const float amask = (m < B_ROWS) ? 1.0f : 0.0f;

  const int kchunk = K >> 2;           // per-wave K range (uniform trip count)
  const int k0 = wave * kchunk;        // scalar

  // Per-lane base pointers at this wave's K offset
  const float* __restrict__ Ap = A + (size_t)(m & (B_ROWS - 1)) * K + k0 + kh;
  const float* __restrict__ Wp = W + (size_t)(k0 + kh) * N + n;

  v8f c = {};
#pragma unroll 4
  for (int i = 0; i < kchunk; i += 4) {
    v2f a, b;
    a.x = Ap[i] * amask;
    a.y = Ap[i + 1] * amask;
    b.x = Wp[(size_t)i * N];
    b.y = Wp[(size_t)(i + 1) * N];
    // D = A(16x4,f32) * B(4x16,f32) + C   -> v_wmma_f32_16x16x4_f32
    c = __builtin_amdgcn_wmma_f32_16x16x4_f32(false, a, false, b,
                                              (short)0, c, false, false);
  }

  // cross-wave reduction of the 4 partial C tiles
  __shared__ float red[4 * 32 * 8];
#pragma unroll
  for (int r = 0; r < 8; ++r) red[(wave * 32 + lane) * 8 + r] = c[r];
  __syncthreads();

  if (wave == 0 && lane < 16) {
    // lanes 0-15 hold M = r (0..7) in C registers r: exactly the valid rows
#pragma unroll
    for (int r = 0; r < 8; ++r) {
      float v = red[(0 * 32 + lane) * 8 + r] + red[(1 * 32 + lane) * 8 + r] +
                red[(2 * 32 + lane) * 8 + r] + red[(3 * 32 + lane) * 8 + r];
      v += bias[n];
      if (RELU) v = fmaxf(v, 0.0f);
      if (TRANSPOSED)
        out[(size_t)n * B_ROWS + r] = v;      // logitsT[k=n][b=r]
      else
        out[(size_t)r * N + n] = v;           // out[b=r][n]
    }
  }
}

// out[b][o] = max_k logitsT[k][b] * CM[o][k]; one block per CM row o.
__global__ __launch_bounds__(256) void tropical_max(const float* __restrict__ ltr,
                                                    const float* __restrict__ CM,
                                                    float* __restrict__ out) {
  const int o = blockIdx.x;
  const float* __restrict__ row = CM + (size_t)o * OUTD;

  float acc[B_ROWS];
#pragma unroll
  for (int r = 0; r < B_ROWS; ++r) acc[r] = -INFINITY;

  // 5120 = 1024 * 5 ; each thread does 5 float4 CM loads
#pragma unroll
  for (int it = 0; it < OUTD / 1024; ++it) {
    const int k = (threadIdx.x + it * 256) * 4;
    const float4 c4 = *(const float4*)(row + k);
    const float ck[4] = {c4.x, c4.y, c4.z, c4.w};
#pragma unroll
    for (int j = 0; j < 4; ++j) {
      const float4 l0 = *(const float4*)(ltr + (size_t)(k + j) * B_ROWS);
      const float4 l1 = *(const float4*)(ltr + (size_t)(k + j) * B_ROWS + 4);
      acc[0] = fmaxf(acc[0], ck[j] * l0.x);
      acc[1] = fmaxf(acc[1], ck[j] * l0.y);
      acc[2] = fmaxf(acc[2], ck[j] * l0.z);
      acc[3] = fmaxf(acc[3], ck[j] * l0.w);
      acc[4] = fmaxf(acc[4], ck[j] * l1.x);
      acc[5] = fmaxf(acc[5], ck[j] * l1.y);
      acc[6] = fmaxf(acc[6], ck[j] * l1.z);
      acc[7] = fmaxf(acc[7], ck[j] * l1.w);
    }
  }

  __shared__ float sm[256 * B_ROWS];
#pragma unroll
  for (int r = 0; r < B_ROWS; ++r) sm[threadIdx.x * B_ROWS + r] = acc[r];
  __syncthreads();
  for (int s = 128; s > 0; s >>= 1) {
    if (threadIdx.x < (unsigned)s) {
#pragma unroll
      for (int r = 0; r < B_ROWS; ++r)
        sm[threadIdx.x * B_ROWS + r] =
            fmaxf(sm[threadIdx.x * B_ROWS + r], sm[(threadIdx.x + s) * B_ROWS + r]);
    }
    __syncthreads();
  }
  if (threadIdx.x < B_ROWS)
    out[(size_t)threadIdx.x * OUTD + o] = sm[threadIdx.x];
}

extern "C" void kernel_launch(void* const* d_in, const int* in_sizes, int n_in,
                              void* d_out, int out_size, void* d_ws, size_t ws_size,
                              hipStream_t stream) {
  const float* x  = (const float*)d_in[0];
  const float* W1 = (const float*)d_in[1];
  const float* b1 = (const float*)d_in[2];
  const float* W2 = (const float*)d_in[3];
  const float* b2 = (const float*)d_in[4];
  const float* W3 = (const float*)d_in[5];
  const float* b3 = (const float*)d_in[6];
  const float* CM = (const float*)d_in[7];

  float* out   = (float*)d_out;                 // [8, 5120]
  float* feats = out + B_ROWS * OUTD;           // [8, 1024] (second return value)

  float* h1  = (float*)d_ws;                    // [8, 2048]   64 KB
  float* ltr = h1 + B_ROWS * H1D;               // [5120, 8]  160 KB (transposed logits)

  // h1 = relu(x @ W1 + b1)
  gemm_wmma_f32<1, 0><<<H1D / 16, 128, 0, stream>>>(x, W1, b1, h1, D_IN, H1D);
  // feats = relu(h1 @ W2 + b2)   (written straight into d_out features slice)
  gemm_wmma_f32<1, 0><<<H2D / 16, 128, 0, stream>>>(h1, W2, b2, feats, H1D, H2D);
  // logitsT = (feats @ W3 + b3)^T
  gemm_wmma_f32<0, 1><<<OUTD / 16, 128, 0, stream>>>(feats, W3, b3, ltr, H2D, OUTD);
  // out[b][o] = max_k logits[b][k] * CM[o][k]
  tropical_max<<<OUTD, 256, 0, stream>>>(ltr, CM, out);
}